// Net_54382875902698
// MI455X (gfx1250) — compile-verified
//
#include <hip/hip_runtime.h>
#include <math.h>

typedef float v2f __attribute__((ext_vector_type(2)));
typedef float v8f __attribute__((ext_vector_type(8)));

static __device__ __forceinline__ float elu1(float v) {
    return v > 0.0f ? v : (__expf(v) - 1.0f);
}

// ---------------- elementwise / edge kernels ----------------

__global__ void zero_f32_kernel(float* __restrict__ p, int n) {
    int i = blockIdx.x * blockDim.x + threadIdx.x;
    if (i < n) p[i] = 0.0f;
}

__global__ void deg_scatter_kernel(const int* __restrict__ col,
                                   const float* __restrict__ w,
                                   float* __restrict__ deg, int E) {
    int e = blockIdx.x * blockDim.x + threadIdx.x;
    if (e < E) atomicAdd(&deg[col[e]], w[e]);
}

__global__ void dinv_kernel(float* __restrict__ deg, int N) {
    int i = blockIdx.x * blockDim.x + threadIdx.x;
    if (i < N) {
        float d = deg[i];
        deg[i] = d > 0.0f ? rsqrtf(d) : 0.0f;
    }
}

__global__ void norm_kernel(const int* __restrict__ row, const int* __restrict__ col,
                            const float* __restrict__ w, const float* __restrict__ dinv,
                            float* __restrict__ norm, int E) {
    int e = blockIdx.x * blockDim.x + threadIdx.x;
    if (e < E) norm[e] = dinv[row[e]] * w[e] * dinv[col[e]];
}

// h[n,f] = elu(x[n] * preW[f] + preb[f]);  F = 64
__global__ void pre_kernel(const float* __restrict__ x, const float* __restrict__ preW,
                           const float* __restrict__ preb, float* __restrict__ h, int total) {
    int i = blockIdx.x * blockDim.x + threadIdx.x;
    if (i >= total) return;
    int n = i >> 6, f = i & 63;
    h[i] = elu1(x[n] * preW[f] + preb[f]);
}

// lane = feature (0..31), wave = edge: coalesced 128B gather of t[row], atomic scatter into agg[col]
__global__ void edge_scatter_kernel(const int* __restrict__ row, const int* __restrict__ col,
                                    const float* __restrict__ norm, const float* __restrict__ t,
                                    float* __restrict__ agg, int E) {
    int gid = blockIdx.x * blockDim.x + threadIdx.x;
    int e = gid >> 5;
    int f = gid & 31;
    if (e >= E) return;
    int r = row[e], c = col[e];
    float v = norm[e] * t[(size_t)r * 32 + f];
    atomicAdd(&agg[(size_t)c * 32 + f], v);
}

// elu(relu(x)) == relu(x)
__global__ void relu_kernel(float* __restrict__ p, int n) {
    int i = blockIdx.x * blockDim.x + threadIdx.x;
    if (i < n) p[i] = fmaxf(p[i], 0.0f);
}

// ---------------- WMMA f32 GEMM ----------------
// One wave computes a 16-row output tile: Ti = A @ Wi  (and Tr = A @ Wr + biasR if DUAL).
// A: [nrows, K] row-major; Wi/Wr: [K, NCT*16] row-major.
// V_WMMA_F32_16X16X4_F32 fragment layouts (ISA 7.12.2):
//   A 16x4:  lane L<16 holds A[M=L][k0], A[M=L][k0+1]; lane L+16 holds K=k0+2,k0+3
//   B 4x16:  mirrored K-split across half-waves, N striped over lanes 0..15
//   C/D:     VGPR r holds M=r (lanes 0-15, N=lane) and M=r+8 (lanes 16-31, N=lane-16)
template <int NCT, bool DUAL>
__global__ void wmma_gemm_kernel(const float* __restrict__ A, int K,
                                 const float* __restrict__ Wi,
                                 const float* __restrict__ Wr,
                                 const float* __restrict__ biasR,
                                 float* __restrict__ Ti,
                                 float* __restrict__ Tr,
                                 int nrows) {
    const int F = NCT * 16;
    int wave = (blockIdx.x * blockDim.x + threadIdx.x) >> 5;
    int lane = threadIdx.x & 31;
    int row0 = wave * 16;
    if (row0 >= nrows) return;

    int m  = lane & 15;   // column (B/C/D) or row (A) within half-wave
    int hi = lane >> 4;   // half-wave select

    v8f accI[NCT] = {};
    v8f accR[NCT] = {};

    int ar = row0 + m;
    if (ar >= nrows) ar = nrows - 1;           // safe for partial tiles
    const float* arow = A + (size_t)ar * K;

    for (int k0 = 0; k0 < K; k0 += 4) {
        int ka = k0 + 2 * hi;
        v2f a;
        a.x = arow[ka];
        a.y = arow[ka + 1];
#pragma unroll
        for (int ct = 0; ct < NCT; ++ct) {
            int colc = ct * 16 + m;
            v2f b;
            b.x = Wi[(size_t)ka * F + colc];
            b.y = Wi[(size_t)(ka + 1) * F + colc];
            accI[ct] = __builtin_amdgcn_wmma_f32_16x16x4_f32(
                false, a, false, b, (short)0, accI[ct], false, false);
            if (DUAL) {
                v2f br;
                br.x = Wr[(size_t)ka * F + colc];
                br.y = Wr[(size_t)(ka + 1) * F + colc];
                accR[ct] = __builtin_amdgcn_wmma_f32_16x16x4_f32(
                    false, a, false, br, (short)0, accR[ct], false, false);
            }
        }
    }

#pragma unroll
    for (int ct = 0; ct < NCT; ++ct) {
        int colc = ct * 16 + m;
        float bv = biasR ? biasR[colc] : 0.0f;
#pragma unroll
        for (int r = 0; r < 8; ++r) {
            int rowm = row0 + r + 8 * hi;
            if (rowm >= nrows) continue;
            size_t idx = (size_t)rowm * F + colc;
            if (DUAL) {
                Ti[idx] = accI[ct][r];
                Tr[idx] = accR[ct][r] + bv;
            } else {
                Ti[idx] = accI[ct][r] + bv;
            }
        }
    }
}

// ---------------- cosine similarity over 16-dim embeddings ----------------

__global__ void score_kernel(const int* __restrict__ li, const float* __restrict__ z,
                             float* __restrict__ out, int L) {
    int e = blockIdx.x * blockDim.x + threadIdx.x;
    if (e >= L) return;
    int i = li[e], j = li[L + e];
    const float4* za = (const float4*)(z + (size_t)i * 16);
    const float4* zb = (const float4*)(z + (size_t)j * 16);
    float dot = 0.0f, na = 0.0f, nb = 0.0f;
#pragma unroll
    for (int q = 0; q < 4; ++q) {
        float4 a = za[q], b = zb[q];
        dot += a.x * b.x + a.y * b.y + a.z * b.z + a.w * b.w;
        na  += a.x * a.x + a.y * a.y + a.z * a.z + a.w * a.w;
        nb  += b.x * b.x + b.y * b.y + b.z * b.z + b.w * b.w;
    }
    float den = fmaxf(sqrtf(na) * sqrtf(nb), 1e-6f);
    out[e] = dot / den;
}

// ---------------- launcher ----------------

static inline int cdiv(long long a, long long b) { return (int)((a + b - 1) / b); }

extern "C" void kernel_launch(void* const* d_in, const int* in_sizes, int n_in,
                              void* d_out, int out_size, void* d_ws, size_t ws_size,
                              hipStream_t stream) {
    const float* x     = (const float*)d_in[0];
    const int*   ei    = (const int*)d_in[1];     // [2, E]
    const float* ew    = (const float*)d_in[2];   // [E]
    const int*   eli   = (const int*)d_in[3];     // [2, L]
    const float* preW  = (const float*)d_in[4];   // [1, 64]
    const float* preb  = (const float*)d_in[5];   // [64]
    const float* g1Wi  = (const float*)d_in[6];   // [64, 32]
    const float* g1Wr  = (const float*)d_in[7];   // [64, 32]
    const float* g1b   = (const float*)d_in[8];   // [32]
    const float* g2Wi  = (const float*)d_in[9];   // [32, 32]
    const float* g2Wr  = (const float*)d_in[10];  // [32, 32]
    const float* g2b   = (const float*)d_in[11];  // [32]
    const float* poW   = (const float*)d_in[12];  // [32, 16]
    const float* pob   = (const float*)d_in[13];  // [16]
    float* out = (float*)d_out;

    const int N = in_sizes[0];       // 100000 (x is [N,1])
    const int E = in_sizes[2];       // 3.2M
    const int L = in_sizes[3] / 2;   // 500K

    const int* erow = ei;
    const int* ecol = ei + E;

    // workspace partition (all f32)
    float* ws   = (float*)d_ws;
    float* deg  = ws;                        // N   (becomes dinv)
    float* norm = deg  + N;                  // E
    float* h64  = norm + E;                  // N*64
    float* t    = h64  + (size_t)N * 64;     // N*32 (A@Wi message source, reused)
    float* agg1 = t    + (size_t)N * 32;     // N*32 (conv1 agg -> h after relu)
    float* agg2 = agg1 + (size_t)N * 32;     // N*32 (conv2 agg -> h after relu)
    float* z    = agg2 + (size_t)N * 32;     // N*16

    const int B = 256;
    const int gemm_threads = cdiv(N, 16) * 32;   // one wave (32 lanes) per 16-row tile

    // gcn_norm
    zero_f32_kernel<<<cdiv(N, B), B, 0, stream>>>(deg, N);
    deg_scatter_kernel<<<cdiv(E, B), B, 0, stream>>>(ecol, ew, deg, E);
    dinv_kernel<<<cdiv(N, B), B, 0, stream>>>(deg, N);
    norm_kernel<<<cdiv(E, B), B, 0, stream>>>(erow, ecol, ew, deg, norm, E);

    // pre: h64 = elu(x @ pre_W + pre_b)
    pre_kernel<<<cdiv((long long)N * 64, B), B, 0, stream>>>(x, preW, preb, h64, N * 64);

    // conv1: t = h64@Wi ; agg1 = h64@Wr + b ; scatter ; relu
    wmma_gemm_kernel<2, true><<<cdiv(gemm_threads, B), B, 0, stream>>>(
        h64, 64, g1Wi, g1Wr, g1b, t, agg1, N);
    edge_scatter_kernel<<<cdiv((long long)E * 32, B), B, 0, stream>>>(
        erow, ecol, norm, t, agg1, E);
    relu_kernel<<<cdiv((long long)N * 32, B), B, 0, stream>>>(agg1, N * 32);

    // conv2: t = h1@Wi ; agg2 = h1@Wr + b ; scatter ; relu
    wmma_gemm_kernel<2, true><<<cdiv(gemm_threads, B), B, 0, stream>>>(
        agg1, 32, g2Wi, g2Wr, g2b, t, agg2, N);
    edge_scatter_kernel<<<cdiv((long long)E * 32, B), B, 0, stream>>>(
        erow, ecol, norm, t, agg2, E);
    relu_kernel<<<cdiv((long long)N * 32, B), B, 0, stream>>>(agg2, N * 32);

    // post: z = h2 @ post_W + post_b
    wmma_gemm_kernel<1, false><<<cdiv(gemm_threads, B), B, 0, stream>>>(
        agg2, 32, poW, nullptr, pob, z, nullptr, N);

    // cosine similarity on label edges
    score_kernel<<<cdiv(L, B), B, 0, stream>>>(eli, z, out, L);
}